// Mp_encoder_46694884442571
// MI455X (gfx1250) — compile-verified
//
#include <hip/hip_runtime.h>
#include <hip/hip_bf16.h>

typedef __attribute__((ext_vector_type(2))) float v2f;
typedef __attribute__((ext_vector_type(8))) float v8f;

#define H 64
#define PNUM 3

// ---------------------------------------------------------------------------
// WMMA helper: D = A(16x4 f32) * B(4x16 f32) + C(16x16 f32)
// ---------------------------------------------------------------------------
__device__ __forceinline__ v8f wmma_f32x4(v2f a, v2f b, v8f c) {
    return __builtin_amdgcn_wmma_f32_16x16x4_f32(
        /*neg_a=*/false, a, /*neg_b=*/false, b,
        /*c_mod=*/(short)0, c, /*reuse_a=*/false, /*reuse_b=*/false);
}

// ---------------------------------------------------------------------------
// OUT[n,k] = sum_j X[n,j] * Wm[k,j]   (X: [N,64], Wm: [64,64] row-major)
// One wave computes a 16-row stripe (all 64 output columns).
// ---------------------------------------------------------------------------
__global__ __launch_bounds__(256) void gemm_xWt_kernel(
    const float* __restrict__ X, const float* __restrict__ Wm,
    float* __restrict__ OUT, int N)
{
    const int wave = blockIdx.x * (blockDim.x >> 5) + (threadIdx.x >> 5);
    const int lane = threadIdx.x & 31;
    const int m    = lane & 15;      // M / N index inside tile
    const int half = lane >> 4;      // selects K-pair
    const int r0   = wave * 16;
    if (r0 + 16 > N) return;

    // Preload A fragments for all 16 K-steps (whole 16x64 stripe of X).
    const float* xrow = X + (size_t)(r0 + m) * H + 2 * half;
    v2f a[16];
#pragma unroll
    for (int jt = 0; jt < 16; ++jt)
        a[jt] = *(const v2f*)(xrow + 4 * jt);

#pragma unroll
    for (int kt = 0; kt < 4; ++kt) {
        const float* wrow = Wm + (size_t)(kt * 16 + m) * H + 2 * half;
        v8f c = {};
#pragma unroll
        for (int jt = 0; jt < 16; ++jt) {
            v2f bfrag = *(const v2f*)(wrow + 4 * jt);
            c = wmma_f32x4(a[jt], bfrag, c);
        }
        // C/D layout: lane holds (M = v + 8*half, N = m)
        float* o = OUT + (size_t)(r0 + 8 * half) * H + kt * 16 + m;
#pragma unroll
        for (int v = 0; v < 8; ++v)
            o[(size_t)v * H] = c[v];
    }
}

// ---------------------------------------------------------------------------
// Attention pre-reduction: Sout[k] += sum_rows tanh( (X @ Wm^T)[n,k] + bias[k] )
// Same WMMA tiling, tanh+row-sum epilogue, atomic accumulation into S[64].
// ---------------------------------------------------------------------------
__global__ __launch_bounds__(256) void gemm_tanh_mean_kernel(
    const float* __restrict__ X, const float* __restrict__ Wm,
    const float* __restrict__ bias, float* __restrict__ Sout, int N)
{
    const int wave = blockIdx.x * (blockDim.x >> 5) + (threadIdx.x >> 5);
    const int lane = threadIdx.x & 31;
    const int m    = lane & 15;
    const int half = lane >> 4;
    const int r0   = wave * 16;
    if (r0 + 16 > N) return;

    const float* xrow = X + (size_t)(r0 + m) * H + 2 * half;
    v2f a[16];
#pragma unroll
    for (int jt = 0; jt < 16; ++jt)
        a[jt] = *(const v2f*)(xrow + 4 * jt);

#pragma unroll
    for (int kt = 0; kt < 4; ++kt) {
        const float* wrow = Wm + (size_t)(kt * 16 + m) * H + 2 * half;
        v8f c = {};
#pragma unroll
        for (int jt = 0; jt < 16; ++jt) {
            v2f bfrag = *(const v2f*)(wrow + 4 * jt);
            c = wmma_f32x4(a[jt], bfrag, c);
        }
        const float bb = bias[kt * 16 + m];
        float sum = 0.0f;
#pragma unroll
        for (int v = 0; v < 8; ++v)
            sum += tanhf(c[v] + bb);
        // combine the two wave halves (same N, different 8-row groups)
        sum += __shfl_xor(sum, 16, 32);
        if (half == 0)
            atomicAdd(&Sout[kt * 16 + m], sum);
    }
}

// ---------------------------------------------------------------------------
// SpMM scatter: agg[row[e], :] += fts[col[e], :] * vals[e]
// One wave per edge; each lane owns a float2 (2*32 = 64 = H).
// ---------------------------------------------------------------------------
__global__ __launch_bounds__(256) void spmm_kernel(
    const float* __restrict__ fts, const float* __restrict__ vals,
    const int* __restrict__ row, const int* __restrict__ col,
    float* __restrict__ agg, int E)
{
    const int lane = threadIdx.x & 31;
    const int wave = blockIdx.x * (blockDim.x >> 5) + (threadIdx.x >> 5);
    const int nwav = gridDim.x * (blockDim.x >> 5);
    for (int e = wave; e < E; e += nwav) {
        const int   cc = col[e];
        const int   rr = row[e];
        const float v  = vals[e];
        const float2 f = *((const float2*)(fts + (size_t)cc * H) + lane);
        float* dst = agg + (size_t)rr * H + 2 * lane;
        atomicAdd(dst,     f.x * v);
        atomicAdd(dst + 1, f.y * v);
    }
}

// ---------------------------------------------------------------------------
// Elementwise kernels
// ---------------------------------------------------------------------------
__global__ void fill_kernel(float* __restrict__ p, int n) {
    for (int i = blockIdx.x * blockDim.x + threadIdx.x; i < n;
         i += gridDim.x * blockDim.x)
        p[i] = 0.0f;
}

__global__ void mask_kernel(const float* __restrict__ x,
                            const unsigned char* __restrict__ mask,
                            float* __restrict__ out, int NHtot) {
    for (int i = blockIdx.x * blockDim.x + threadIdx.x; i < NHtot;
         i += gridDim.x * blockDim.x)
        out[i] = mask[i >> 6] ? 0.0f : x[i];
}

__global__ void bias_prelu_kernel(const float* __restrict__ agg,
                                  const float* __restrict__ b,
                                  const float* __restrict__ a, int p,
                                  float* __restrict__ out, int NHtot) {
    const float slope = a[p];
    for (int i = blockIdx.x * blockDim.x + threadIdx.x; i < NHtot;
         i += gridDim.x * blockDim.x) {
        float o = agg[i] + b[i & (H - 1)];
        out[i] = o > 0.0f ? o : slope * o;
    }
}

// beta = softmax_p( (S[p,:]/N) . att )   — S holds column sums of tanh(...)
__global__ void beta_kernel(const float* __restrict__ S,
                            const float* __restrict__ att,
                            float* __restrict__ beta, float invN) {
    const int lane = threadIdx.x;  // 32 threads
    float w[PNUM];
#pragma unroll
    for (int p = 0; p < PNUM; ++p) {
        float acc = S[p * H + lane] * att[lane] +
                    S[p * H + lane + 32] * att[lane + 32];
#pragma unroll
        for (int o = 16; o > 0; o >>= 1)
            acc += __shfl_xor(acc, o, 32);
        w[p] = acc * invN;
    }
    float mx = fmaxf(w[0], fmaxf(w[1], w[2]));
    float e0 = __expf(w[0] - mx), e1 = __expf(w[1] - mx), e2 = __expf(w[2] - mx);
    float s = e0 + e1 + e2;
    if (lane == 0) {
        beta[0] = e0 / s;
        beta[1] = e1 / s;
        beta[2] = e2 / s;
    }
}

__global__ void combine_kernel(const float* __restrict__ EMB,
                               const float* __restrict__ beta,
                               float* __restrict__ z, int NHtot) {
    const float b0 = beta[0], b1 = beta[1], b2 = beta[2];
    const size_t nh = (size_t)NHtot;
    for (int i = blockIdx.x * blockDim.x + threadIdx.x; i < NHtot;
         i += gridDim.x * blockDim.x)
        z[i] = b0 * EMB[i] + b1 * EMB[nh + i] + b2 * EMB[2 * nh + i];
}

// ---------------------------------------------------------------------------
// Host launcher
// ---------------------------------------------------------------------------
extern "C" void kernel_launch(void* const* d_in, const int* in_sizes, int n_in,
                              void* d_out, int out_size, void* d_ws, size_t ws_size,
                              hipStream_t stream) {
    const float*         h     = (const float*)d_in[0];
    const float*         W     = (const float*)d_in[1];
    const float*         b     = (const float*)d_in[2];
    const float*         a     = (const float*)d_in[3];
    const float*         fc_w  = (const float*)d_in[4];
    const float*         fc_b  = (const float*)d_in[5];
    const float*         att   = (const float*)d_in[6];
    const float*         adj   = (const float*)d_in[7];
    const int*           row   = (const int*)d_in[8];
    const int*           col   = (const int*)d_in[9];
    const unsigned char* mask1 = (const unsigned char*)d_in[10];
    const unsigned char* mask2 = (const unsigned char*)d_in[11];
    float*               out   = (float*)d_out;

    const int    N   = in_sizes[0] / H;   // 100000
    const int    E   = in_sizes[7];       // 1600000
    const size_t NHs = (size_t)N * H;
    const int    NHi = (int)NHs;

    float* ws   = (float*)d_ws;
    float* X    = ws;                 // [N,H]
    float* FTS  = ws + NHs;           // [N,H]
    float* AGG  = ws + 2 * NHs;       // [N,H]
    float* EMB  = ws + 3 * NHs;       // [P,N,H]
    float* S    = ws + 6 * NHs;       // [P,64] (+ pad)
    float* BETA = S + 256;            // [P]

    const int gemmBlocks = (N / 16 + 7) / 8;  // 8 waves per 256-thread block
    const int ewBlocks   = 2048;
    const int spmmBlocks = 4096;

    auto pipeline = [&](const float* Xin, float* z) {
        for (int p = 0; p < PNUM; ++p) {
            gemm_xWt_kernel<<<gemmBlocks, 256, 0, stream>>>(
                Xin, W + (size_t)p * H * H, FTS, N);
            fill_kernel<<<1024, 256, 0, stream>>>(AGG, NHi);
            spmm_kernel<<<spmmBlocks, 256, 0, stream>>>(FTS, adj, row, col, AGG, E);
            bias_prelu_kernel<<<ewBlocks, 256, 0, stream>>>(
                AGG, b + p * H, a, p, EMB + (size_t)p * NHs, NHi);
        }
        fill_kernel<<<1, 256, 0, stream>>>(S, PNUM * H);
        for (int p = 0; p < PNUM; ++p)
            gemm_tanh_mean_kernel<<<gemmBlocks, 256, 0, stream>>>(
                EMB + (size_t)p * NHs, fc_w, fc_b, S + p * H, N);
        beta_kernel<<<1, 32, 0, stream>>>(S, att, BETA, 1.0f / (float)N);
        combine_kernel<<<ewBlocks, 256, 0, stream>>>(EMB, BETA, z, NHi);
    };

    // z_mp = attention(gcn_all(h * ~mask1))
    mask_kernel<<<ewBlocks, 256, 0, stream>>>(h, mask1, X, NHi);
    pipeline(X, out);

    // z_mp2 = attention(gcn_all(h))
    pipeline(h, out + NHs);

    // x_recon = attention(gcn_all(z_mp * ~mask2))
    mask_kernel<<<ewBlocks, 256, 0, stream>>>(out, mask2, X, NHi);
    pipeline(X, out + 2 * NHs);
}